// MultiPositionTransfer_9053791060240
// MI455X (gfx1250) — compile-verified
//
#include <hip/hip_runtime.h>

// Problem constants from the reference: T=256, B=8, D=256, N=64
#define TLEN  256
#define BATCH 8
#define DEMB  256
#define NPOS  64
#define NROWS (TLEN * BATCH)   // 2048 (i,b) row-pairs

typedef __attribute__((ext_vector_type(2))) float v2f;
typedef __attribute__((ext_vector_type(8))) float v8f;

// ---------------------------------------------------------------------------
// Kernel 1: zero the per-position bin counters (d_ws is poisoned, not zeroed).
// ---------------------------------------------------------------------------
__global__ void zero_counts_kernel(int* __restrict__ counts) {
    if (threadIdx.x < NPOS) counts[threadIdx.x] = 0;
}

// ---------------------------------------------------------------------------
// Kernel 2: bin rows by position value. idx[n*NROWS + slot] = row.
// Slot order is nondeterministic (atomics) but each row's output is
// independent of its slot, so d_out is deterministic.
// ---------------------------------------------------------------------------
__global__ void bin_rows_kernel(const long long* __restrict__ positions,
                                int* __restrict__ counts,
                                int* __restrict__ idx) {
    const int r = blockIdx.x * blockDim.x + threadIdx.x;
    if (r < NROWS) {
        const int n = (int)positions[r];          // values in [0, 64)
        const int slot = atomicAdd(&counts[n], 1);
        idx[n * NROWS + slot] = r;
    }
}

// ---------------------------------------------------------------------------
// Scatter one 16x16 C/D tile. The 8 row indices a lane needs are contiguous
// (rbase..rbase+7), so fetch them as two b128 loads up front; fully-valid
// tiles take an unguarded store path.
// C/D layout: VGPR j -> (M = j + 8*khalf, N = lane&15).
// ---------------------------------------------------------------------------
__device__ __forceinline__ void store_tile(const v8f& acc, int mt, int khalf,
                                           int col, const int* __restrict__ bin,
                                           int cnt, float* __restrict__ out) {
    const int rbase = mt * 16 + 8 * khalf;                 // multiple of 8
    const int4* bp = (const int4*)(bin + rbase);           // 32B-aligned
    const int4 b0 = bp[0];
    const int4 b1 = bp[1];
    if (rbase + 8 <= cnt) {                                // fast path: full half-tile
        out[(size_t)b0.x * DEMB + col] = acc[0];
        out[(size_t)b0.y * DEMB + col] = acc[1];
        out[(size_t)b0.z * DEMB + col] = acc[2];
        out[(size_t)b0.w * DEMB + col] = acc[3];
        out[(size_t)b1.x * DEMB + col] = acc[4];
        out[(size_t)b1.y * DEMB + col] = acc[5];
        out[(size_t)b1.z * DEMB + col] = acc[6];
        out[(size_t)b1.w * DEMB + col] = acc[7];
    } else {                                               // ragged tail
        const int rows[8] = {b0.x, b0.y, b0.z, b0.w, b1.x, b1.y, b1.z, b1.w};
        #pragma unroll
        for (int j = 0; j < 8; ++j)
            if (rbase + j < cnt) out[(size_t)rows[j] * DEMB + col] = acc[j];
    }
}

// ---------------------------------------------------------------------------
// Kernel 3: per-position GEMM  Out_bin[M x 256] = A_bin[M x 256] * table[n].
// grid = (NPOS, 4 column-groups of 64), block = 128 threads = 4 waves.
// Each wave owns a 16-wide column slice and processes TWO 16-row M-tiles at
// a time with dual accumulators sharing one B-fragment load (halves table
// traffic and B-load issue per WMMA). K=256 via 2x64 v_wmma_f32_16x16x4_f32
// per tile-pair step.
// ---------------------------------------------------------------------------
__global__ void __launch_bounds__(128)
gemm_binned_kernel(const float* __restrict__ outputs,   // [NROWS, DEMB]
                   const float* __restrict__ table,     // [NPOS, DEMB, DEMB]
                   const int*   __restrict__ counts,
                   const int*   __restrict__ idx,
                   float*       __restrict__ out) {     // [NROWS, DEMB]
    const int n     = blockIdx.x;
    const int wave  = threadIdx.x >> 5;     // 0..3 (wave32)
    const int lane  = threadIdx.x & 31;
    const int m16   = lane & 15;
    const int khalf = lane >> 4;            // 0: K+{0,1}, 1: K+{2,3} (ISA A/B layout)
    const int col   = blockIdx.y * 64 + wave * 16 + m16;

    const int  cnt = counts[n];
    const int* bin = idx + n * NROWS;
    const float* tb = table + (size_t)n * DEMB * DEMB + col;  // column base

    // Loop bound is uniform per block: EXEC is all-ones around every WMMA.
    for (int mt0 = 0; mt0 * 16 < cnt; mt0 += 2) {
        const int mt1 = mt0 + 1;
        const int rl0 = mt0 * 16 + m16;
        const int rl1 = mt1 * 16 + m16;
        const bool v0 = rl0 < cnt;
        const bool v1 = rl1 < cnt;
        // cnt > 0 inside the loop, so bin[0] is a safe fallback row.
        const int g0 = v0 ? bin[rl0] : bin[0];
        const int g1 = v1 ? bin[rl1] : bin[0];
        const float* __restrict__ a0p = outputs + (size_t)g0 * DEMB;
        const float* __restrict__ a1p = outputs + (size_t)g1 * DEMB;

        v8f acc0 = {0.f, 0.f, 0.f, 0.f, 0.f, 0.f, 0.f, 0.f};
        v8f acc1 = {0.f, 0.f, 0.f, 0.f, 0.f, 0.f, 0.f, 0.f};

        if (mt0 * 16 + 32 <= cnt) {
            // Common case: both tiles fully valid -> no masking VALU at all.
            #pragma unroll 8
            for (int k0 = 0; k0 < DEMB; k0 += 4) {
                const int ka = k0 + 2 * khalf;
                v2f b;                                   // B: (K=ka..ka+1, N=col)
                b.x = tb[(size_t)ka       * DEMB];
                b.y = tb[(size_t)(ka + 1) * DEMB];
                const v2f A0 = *(const v2f*)(a0p + ka);  // A: (M=m16, K=ka..ka+1)
                const v2f A1 = *(const v2f*)(a1p + ka);
                acc0 = __builtin_amdgcn_wmma_f32_16x16x4_f32(
                           false, A0, false, b, (short)0, acc0, false, false);
                acc1 = __builtin_amdgcn_wmma_f32_16x16x4_f32(
                           false, A1, false, b, (short)0, acc1, false, false);
            }
        } else {
            // Ragged tail: zero-pad invalid rows with a multiplicative mask.
            const float msk0 = v0 ? 1.0f : 0.0f;
            const float msk1 = v1 ? 1.0f : 0.0f;
            #pragma unroll 8
            for (int k0 = 0; k0 < DEMB; k0 += 4) {
                const int ka = k0 + 2 * khalf;
                v2f b;
                b.x = tb[(size_t)ka       * DEMB];
                b.y = tb[(size_t)(ka + 1) * DEMB];
                v2f A0 = *(const v2f*)(a0p + ka);
                v2f A1 = *(const v2f*)(a1p + ka);
                A0.x *= msk0; A0.y *= msk0;
                A1.x *= msk1; A1.y *= msk1;
                acc0 = __builtin_amdgcn_wmma_f32_16x16x4_f32(
                           false, A0, false, b, (short)0, acc0, false, false);
                acc1 = __builtin_amdgcn_wmma_f32_16x16x4_f32(
                           false, A1, false, b, (short)0, acc1, false, false);
            }
        }

        store_tile(acc0, mt0, khalf, col, bin, cnt, out);
        if (mt1 * 16 < cnt)                              // avoid OOB bin reads
            store_tile(acc1, mt1, khalf, col, bin, cnt, out);
    }
}

// ---------------------------------------------------------------------------
// Host-side launcher. Inputs (setup_inputs order):
//   d_in[0] = positions [T,B]   int64
//   d_in[1] = outputs   [T,B,D] fp32
//   d_in[2] = table     [N,D,D] fp32
// d_out = [T,B,D] fp32.  d_ws layout: counts[64] ints, then idx[64*2048] ints.
// ---------------------------------------------------------------------------
extern "C" void kernel_launch(void* const* d_in, const int* in_sizes, int n_in,
                              void* d_out, int out_size, void* d_ws, size_t ws_size,
                              hipStream_t stream) {
    const long long* positions = (const long long*)d_in[0];
    const float*     outputs   = (const float*)d_in[1];
    const float*     table     = (const float*)d_in[2];
    float*           out       = (float*)d_out;

    int* counts = (int*)d_ws;
    int* idx    = counts + NPOS;   // 256-byte offset, 16B-aligned bins

    zero_counts_kernel<<<1, 64, 0, stream>>>(counts);
    bin_rows_kernel<<<(NROWS + 255) / 256, 256, 0, stream>>>(positions, counts, idx);

    dim3 grid(NPOS, DEMB / 64);    // 64 positions x 4 column-groups = 256 blocks
    gemm_binned_kernel<<<grid, 128, 0, stream>>>(outputs, table, counts, idx, out);
}